// Sampler_24086176596564
// MI455X (gfx1250) — compile-verified
//
#include <hip/hip_runtime.h>
#include <stdint.h>

#define VV 128000
#define HALF_ROWS 128
#define HALF_N 16384000u      // 128 * 128000
#define NTHREADS 800          // 25 waves
#define SPLITS 4              // vocab chunks per row-pair -> 512 blocks
#define CHUNK 32000           // 128000 / 4
#define NTILES 10             // 32000 / (800*4)
#define NWAVES 25
#define NLN2 -0.69314718056f

__device__ __forceinline__ uint32_t rotl32(uint32_t x, uint32_t r) {
  return (x << r) | (x >> (32u - r));
}

// JAX Threefry-2x32 with key = jax.random.key(1) -> (k0,k1)=(0,1)
__device__ __forceinline__ void threefry2x32(uint32_t x0, uint32_t x1,
                                             uint32_t& o0, uint32_t& o1) {
  const uint32_t ks0 = 0u, ks1 = 1u, ks2 = 0x1BD11BDBu; // 0^1^0x1BD11BDA
  x0 += ks0; x1 += ks1;
#define TF_R4(a,b,c,d)                                   \
  x0 += x1; x1 = rotl32(x1,(a)); x1 ^= x0;               \
  x0 += x1; x1 = rotl32(x1,(b)); x1 ^= x0;               \
  x0 += x1; x1 = rotl32(x1,(c)); x1 ^= x0;               \
  x0 += x1; x1 = rotl32(x1,(d)); x1 ^= x0;
  TF_R4(13,15,26,6)   x0 += ks1; x1 += ks2 + 1u;
  TF_R4(17,29,16,24)  x0 += ks2; x1 += ks0 + 2u;
  TF_R4(13,15,26,6)   x0 += ks0; x1 += ks1 + 3u;
  TF_R4(17,29,16,24)  x0 += ks1; x1 += ks2 + 4u;
  TF_R4(13,15,26,6)   x0 += ks2; x1 += ks0 + 5u;
#undef TF_R4
  o0 = x0; o1 = x1;
}

// bits -> f in [1,2); noise = -log1p(-u) computed as (-ln2)*log2(2-f), the
// argument 2-f being exact in fp32.  val = logit + t*(-log(noise)) folded to
// fma(-t*ln2, log2(noise), logit).  u==0 -> noise=-0 -> log2=-inf -> val=+inf
// (matches reference probs/0 = inf); t==0 rows give NaN, discarded (greedy).
__device__ __forceinline__ float sample_val(uint32_t bits, float tln2, float logit) {
  float f = __uint_as_float((bits >> 9) | 0x3f800000u);
  float noise = NLN2 * __log2f(2.0f - f);
  return fmaf(tln2, __log2f(noise), logit);
}

__device__ __forceinline__ void async_ld_f4(uint32_t lds_addr, uint32_t voff,
                                            unsigned long long saddr) {
  // CDNA5 async copy global -> LDS, tracked on ASYNCcnt (per-lane 16B)
  asm volatile("global_load_async_to_lds_b128 %0, %1, %2"
               :: "v"(lds_addr), "v"(voff), "s"(saddr) : "memory");
}

__device__ __forceinline__ void amax_upd(float& v, int& i, float ov, int oi) {
  if (ov > v || (ov == v && oi < i)) { v = ov; i = oi; } // lowest-index ties
}

__device__ __forceinline__ void wave_argmax(float& v, int& i) {
#pragma unroll
  for (int off = 16; off > 0; off >>= 1) {
    float ov = __shfl_down(v, off, 32);
    int   oi = __shfl_down(i, off, 32);
    amax_upd(v, i, ov, oi);
  }
}

__global__ __launch_bounds__(NTHREADS)
void sampler_partial_kernel(const float* __restrict__ logits,
                            const float* __restrict__ temps,
                            float* __restrict__ gv, int* __restrict__ gi,
                            float* __restrict__ sv, int* __restrict__ si) {
  __shared__ float4 bufA[2][NTHREADS];   // rows r     , double buffered
  __shared__ float4 bufB[2][NTHREADS];   // rows r+128 , double buffered
  __shared__ float rv[4][32];
  __shared__ int   ri[4][32];

  const int t     = threadIdx.x;
  const int rowA  = blockIdx.x & (HALF_ROWS - 1);
  const int split = blockIdx.x >> 7;
  const int rowB  = rowA + HALF_ROWS;
  const int v0    = split * CHUNK;                    // first vocab index
  const unsigned long long baseA =
      (unsigned long long)(uintptr_t)(logits + (size_t)rowA * VV + v0);
  const unsigned long long baseB =
      (unsigned long long)(uintptr_t)(logits + (size_t)rowB * VV + v0);
  const float tmpA = temps[rowA];
  const float tmpB = temps[rowB];
  const float tln2A = NLN2 * tmpA;                    // -t*ln2
  const float tln2B = NLN2 * tmpB;

  const uint32_t ldsA[2] = { (uint32_t)(uintptr_t)&bufA[0][t],
                             (uint32_t)(uintptr_t)&bufA[1][t] };
  const uint32_t ldsB[2] = { (uint32_t)(uintptr_t)&bufB[0][t],
                             (uint32_t)(uintptr_t)&bufB[1][t] };

  float gmaxA = -__builtin_inff(), gmaxB = -__builtin_inff();
  float smaxA = -__builtin_inff(), smaxB = -__builtin_inff();
  int gidxA = 0, gidxB = 0, sidxA = 0, sidxB = 0;

  { // prologue: stage tile 0
    uint32_t voff = (uint32_t)t * 16u;
    async_ld_f4(ldsA[0], voff, baseA);
    async_ld_f4(ldsB[0], voff, baseB);
  }

  for (int i = 0; i < NTILES; ++i) {
    const int s = i & 1;
    if (i + 1 < NTILES) {
      uint32_t voff = (uint32_t)((i + 1) * NTHREADS + t) * 16u;
      async_ld_f4(ldsA[s ^ 1], voff, baseA);
      async_ld_f4(ldsB[s ^ 1], voff, baseB);
      // allow next tile's 2 loads in flight; current tile's 2 must be done
      asm volatile("s_wait_asynccnt 2" ::: "memory");
    } else {
      asm volatile("s_wait_asynccnt 0" ::: "memory");
    }
    // each lane reads back exactly the 16B it async-loaded: no barrier needed
    float4 la = bufA[s][t];
    float4 lb = bufB[s][t];
    const int vbase = v0 + (i * NTHREADS + t) * 4;
    const float lav[4] = { la.x, la.y, la.z, la.w };
    const float lbv[4] = { lb.x, lb.y, lb.z, lb.w };
#pragma unroll
    for (int c = 0; c < 4; ++c) {
      const int v = vbase + c;
      const float a = lav[c];
      const float b = lbv[c];
      // flat index of (rowA,v); threefry pairs it with +HALF_N = (rowB,v),
      // so one threefry block supplies bits for both rows.
      uint32_t j = (uint32_t)rowA * (uint32_t)VV + (uint32_t)v;
      uint32_t r0, r1;
      threefry2x32(j, j + HALF_N, r0, r1);
      float va = sample_val(r0, tln2A, a);
      float vb = sample_val(r1, tln2B, b);
      if (a  > gmaxA) { gmaxA = a;  gidxA = v; }
      if (b  > gmaxB) { gmaxB = b;  gidxB = v; }
      if (va > smaxA) { smaxA = va; sidxA = v; }
      if (vb > smaxB) { smaxB = vb; sidxB = v; }
    }
  }

  wave_argmax(gmaxA, gidxA);
  wave_argmax(gmaxB, gidxB);
  wave_argmax(smaxA, sidxA);
  wave_argmax(smaxB, sidxB);

  const int lane = t & 31, wid = t >> 5;
  if (lane == 0) {
    rv[0][wid] = gmaxA; ri[0][wid] = gidxA;
    rv[1][wid] = gmaxB; ri[1][wid] = gidxB;
    rv[2][wid] = smaxA; ri[2][wid] = sidxA;
    rv[3][wid] = smaxB; ri[3][wid] = sidxB;
  }
  __syncthreads();
  if (t < 32) {
    const bool ok = t < NWAVES;
    float w0 = ok ? rv[0][t] : -__builtin_inff(); int i0 = ok ? ri[0][t] : 0;
    float w1 = ok ? rv[1][t] : -__builtin_inff(); int i1 = ok ? ri[1][t] : 0;
    float w2 = ok ? rv[2][t] : -__builtin_inff(); int i2 = ok ? ri[2][t] : 0;
    float w3 = ok ? rv[3][t] : -__builtin_inff(); int i3 = ok ? ri[3][t] : 0;
    wave_argmax(w0, i0);
    wave_argmax(w1, i1);
    wave_argmax(w2, i2);
    wave_argmax(w3, i3);
    if (t == 0) {
      const int kA = rowA * SPLITS + split;
      const int kB = rowB * SPLITS + split;
      gv[kA] = w0; gi[kA] = i0; sv[kA] = w2; si[kA] = i2;
      gv[kB] = w1; gi[kB] = i1; sv[kB] = w3; si[kB] = i3;
    }
  }
}

__global__ __launch_bounds__(256)
void sampler_finalize_kernel(const float* __restrict__ temps,
                             const float* __restrict__ gv, const int* __restrict__ gi,
                             const float* __restrict__ sv, const int* __restrict__ si,
                             int* __restrict__ out) {
  const int r = threadIdx.x;   // one thread per row, 256 threads
  float bgv = -__builtin_inff(); int bgi = 0;
  float bsv = -__builtin_inff(); int bsi = 0;
#pragma unroll
  for (int s = 0; s < SPLITS; ++s) {
    const int k = r * SPLITS + s;
    amax_upd(bgv, bgi, gv[k], gi[k]);
    amax_upd(bsv, bsi, sv[k], si[k]);
  }
  out[r] = (temps[r] == 0.0f) ? bgi : bsi;
}

extern "C" void kernel_launch(void* const* d_in, const int* in_sizes, int n_in,
                              void* d_out, int out_size, void* d_ws, size_t ws_size,
                              hipStream_t stream) {
  const float* logits = (const float*)d_in[0];   // (256, 128000) f32
  const float* temps  = (const float*)d_in[1];   // (256,) f32
  int* out = (int*)d_out;                        // (256,) i32 tokens
  (void)in_sizes; (void)n_in; (void)out_size; (void)ws_size;

  // workspace: 4 arrays of 256*SPLITS entries (16 KB total), fully
  // rewritten every call before being read -> deterministic across replays
  float* gv = (float*)d_ws;
  int*   gi = (int*)(gv + 256 * SPLITS);
  float* sv = (float*)(gi + 256 * SPLITS);
  int*   si = (int*)(sv + 256 * SPLITS);

  sampler_partial_kernel<<<dim3(HALF_ROWS * SPLITS), dim3(NTHREADS), 0, stream>>>(
      logits, temps, gv, gi, sv, si);
  sampler_finalize_kernel<<<dim3(1), dim3(256), 0, stream>>>(
      temps, gv, gi, sv, si, out);
}